// FastenRGAT_63608465653854
// MI455X (gfx1250) — compile-verified
//
#include <hip/hip_runtime.h>
#include <math.h>

#define R_REL 8
#define F_HID 64
#define F_OUT 40
#define F_IN  128

typedef __attribute__((ext_vector_type(2))) float v2f;
typedef __attribute__((ext_vector_type(8))) float v8f;

// Monotone float->int encoding so integer atomicMax == float max.
__device__ __forceinline__ int fenc(float f) {
  int i = __float_as_int(f);
  return (i >= 0) ? i : (i ^ 0x7fffffff);
}
__device__ __forceinline__ float fdec(int i) {
  return __int_as_float((i >= 0) ? i : (i ^ 0x7fffffff));
}

// ---------------------------------------------------------------------------
// Dense transform: OUT[n, r, :] = X[n, :] @ W[r]   (fp32-exact WMMA)
// X:[n_nodes,K] row-major, W:[R,K,F_HID] row-major, OUT:[n_nodes,R,F_HID].
// grid = (n_nodes/16, R), block = 128 (4 waves); wave w -> output cols [16w,16w+16).
// A 16x4 layout: lane l holds row (l&15), K-pair starting at k + 2*(l>>4).
// B 4x16 layout: lane l holds col (l&15), same K-pair, one element per VGPR.
// C/D 16x16:    acc[i] is row i + 8*(l>>4), col (l&15).
// ---------------------------------------------------------------------------
__global__ __launch_bounds__(128)
void xform_wmma(const float* __restrict__ X, const float* __restrict__ W,
                float* __restrict__ OUT, int K) {
  const int m0   = blockIdx.x * 16;
  const int r    = blockIdx.y;
  const int wave = threadIdx.x >> 5;
  const int lane = threadIdx.x & 31;
  const int ml   = lane & 15;
  const int hi   = lane >> 4;
  const int n0   = wave * 16;

  const float* __restrict__ arow = X + (size_t)(m0 + ml) * K;
  const float* __restrict__ Wr   = W + (size_t)r * K * F_HID;

  v8f acc = {};
  for (int k = 0; k < K; k += 4) {
    const int ka = k + 2 * hi;
    v2f a, b;
    a.x = arow[ka];
    a.y = arow[ka + 1];
    b.x = Wr[(size_t)ka * F_HID + n0 + ml];
    b.y = Wr[(size_t)(ka + 1) * F_HID + n0 + ml];
    acc = __builtin_amdgcn_wmma_f32_16x16x4_f32(false, a, false, b,
                                                (short)0, acc, false, false);
  }
#pragma unroll
  for (int i = 0; i < 8; ++i) {
    const int row = m0 + i + 8 * hi;
    OUT[((size_t)row * R_REL + r) * F_HID + n0 + ml] = acc[i];
  }
}

// qn[t] = xw[t,:]·q, kn[t] = xw[t,:]·k   for t in [0, N*R)
__global__ void qk_kernel(const float* __restrict__ xw,
                          const float* __restrict__ q, const float* __restrict__ k,
                          float* __restrict__ qn, float* __restrict__ kn, int total) {
  int t = blockIdx.x * blockDim.x + threadIdx.x;
  if (t >= total) return;
  const float4* row = (const float4*)(xw + (size_t)t * F_HID);
  const float4* qv4 = (const float4*)q;
  const float4* kv4 = (const float4*)k;
  float sq = 0.f, sk = 0.f;
#pragma unroll
  for (int i = 0; i < F_HID / 4; ++i) {
    float4 v = row[i], qq = qv4[i], kk = kv4[i];
    sq += v.x * qq.x + v.y * qq.y + v.z * qq.z + v.w * qq.w;
    sk += v.x * kk.x + v.y * kk.y + v.z * kk.z + v.w * kk.w;
  }
  qn[t] = sq;
  kn[t] = sk;
}

__global__ void init_amax(int* __restrict__ amaxI, int n) {
  int t = blockIdx.x * blockDim.x + threadIdx.x;
  if (t < n) amaxI[t] = fenc(-INFINITY);
}

// Pass 1: alpha = leaky_relu(qn[dst,et] + kn[src,et]); segment-max via int atomics.
__global__ void edge_alpha(const int* __restrict__ src, const int* __restrict__ dst,
                           const int* __restrict__ etype,
                           const float* __restrict__ qn, const float* __restrict__ kn,
                           float* __restrict__ alpha, int* __restrict__ amaxI, int n_edges) {
  int e = blockIdx.x * blockDim.x + threadIdx.x;
  if (e >= n_edges) return;
  int s = src[e], d = dst[e], ty = etype[e];
  float a = qn[d * R_REL + ty] + kn[s * R_REL + ty];
  a = (a > 0.f) ? a : 0.2f * a;
  alpha[e] = a;
  atomicMax(&amaxI[d], fenc(a));
}

__global__ void decode_amax(const int* __restrict__ amaxI, float* __restrict__ amaxF, int n) {
  int t = blockIdx.x * blockDim.x + threadIdx.x;
  if (t >= n) return;
  float f = fdec(amaxI[t]);
  amaxF[t] = isfinite(f) ? f : 0.0f;
}

// Pass 2: ex = exp(alpha - amax[dst]); denom[dst] += ex (overwrites alpha buf).
__global__ void edge_exp(const int* __restrict__ dst, float* __restrict__ alpha,
                         const float* __restrict__ amaxF, float* __restrict__ denom,
                         int n_edges) {
  int e = blockIdx.x * blockDim.x + threadIdx.x;
  if (e >= n_edges) return;
  int d = dst[e];
  float ex = __expf(alpha[e] - amaxF[d]);
  alpha[e] = ex;
  atomicAdd(&denom[d], ex);
}

// Pass 3: one wave32 per edge; lane owns float2 of the 64-wide message.
// xw (102 MB) is L2-resident on MI455X (192 MB L2) -> gathers hit L2.
__global__ void edge_aggregate(const int* __restrict__ src, const int* __restrict__ dst,
                               const int* __restrict__ etype,
                               const float* __restrict__ ex, const float* __restrict__ denom,
                               const float* __restrict__ xw, float* __restrict__ agg,
                               int n_edges) {
  long long t = (long long)blockIdx.x * blockDim.x + threadIdx.x;
  int e = (int)(t >> 5);
  if (e >= n_edges) return;
  int lane = (int)(t & 31);
  int s = src[e], d = dst[e], ty = etype[e];
  const float* mrow = xw + ((size_t)s * R_REL + ty) * F_HID + lane * 2;
  __builtin_prefetch(mrow, 0, 1);
  float attn = ex[e] / (denom[d] + 1e-16f);
  float2 v = *(const float2*)mrow;
  float* o = agg + (size_t)d * F_HID + lane * 2;
  atomicAdd(o, attn * v.x);
  atomicAdd(o + 1, attn * v.y);
}

__global__ void bias_relu(float* __restrict__ h, const float* __restrict__ b, int total) {
  int t = blockIdx.x * blockDim.x + threadIdx.x;
  if (t >= total) return;
  float v = h[t] + b[t & (F_HID - 1)];
  h[t] = (v > 0.f) ? v : 0.f;
}

// logits = h2 @ lin_w + lin_b; out = log_softmax(logits). One thread per node.
__global__ void head_logsoftmax(const float* __restrict__ h2,
                                const float* __restrict__ lw, const float* __restrict__ lb,
                                float* __restrict__ out, int n_nodes) {
  int n = blockIdx.x * blockDim.x + threadIdx.x;
  if (n >= n_nodes) return;
  float hloc[F_HID];
  const float* hr = h2 + (size_t)n * F_HID;
#pragma unroll
  for (int i = 0; i < F_HID; ++i) hloc[i] = hr[i];
  float logits[F_OUT];
  float mx = -INFINITY;
#pragma unroll 4
  for (int o = 0; o < F_OUT; ++o) {
    float s = lb[o];
    for (int i = 0; i < F_HID; ++i) s += hloc[i] * lw[i * F_OUT + o];
    logits[o] = s;
    mx = (s > mx) ? s : mx;
  }
  float sum = 0.f;
  for (int o = 0; o < F_OUT; ++o) sum += __expf(logits[o] - mx);
  float lse = mx + __logf(sum);
  float* orow = out + (size_t)n * F_OUT;
  for (int o = 0; o < F_OUT; ++o) orow[o] = logits[o] - lse;
}

extern "C" void kernel_launch(void* const* d_in, const int* in_sizes, int n_in,
                              void* d_out, int out_size, void* d_ws, size_t ws_size,
                              hipStream_t stream) {
  const float* x   = (const float*)d_in[0];
  const int* eidx  = (const int*)d_in[1];
  const int* etype = (const int*)d_in[2];
  // d_in[3] = tensor_slice (unused metadata)
  const float* w1 = (const float*)d_in[4];
  const float* q1 = (const float*)d_in[5];
  const float* k1 = (const float*)d_in[6];
  const float* b1 = (const float*)d_in[7];
  const float* w2 = (const float*)d_in[8];
  const float* q2 = (const float*)d_in[9];
  const float* k2 = (const float*)d_in[10];
  const float* b2 = (const float*)d_in[11];
  const float* lw = (const float*)d_in[12];
  const float* lb = (const float*)d_in[13];
  float* out = (float*)d_out;

  const int n_nodes = in_sizes[0] / F_IN; // 50000 (divisible by 16)
  const int n_edges = in_sizes[1] / 2;    // 1600000
  const int* src = eidx;
  const int* dst = eidx + n_edges;

  // Workspace layout (256B-aligned slices)
  char* ws = (char*)d_ws;
  size_t off = 0;
  auto alloc = [&](size_t bytes) -> void* {
    void* p = ws + off;
    off = (off + bytes + 255) & ~(size_t)255;
    return p;
  };
  float* xw    = (float*)alloc((size_t)n_nodes * R_REL * F_HID * 4); // 102.4 MB, L2-resident
  float* qn    = (float*)alloc((size_t)n_nodes * R_REL * 4);
  float* kn    = (float*)alloc((size_t)n_nodes * R_REL * 4);
  float* exbuf = (float*)alloc((size_t)n_edges * 4);
  int*   amaxI = (int*)  alloc((size_t)n_nodes * 4);
  float* amaxF = (float*)alloc((size_t)n_nodes * 4);
  float* denom = (float*)alloc((size_t)n_nodes * 4);
  float* h1    = (float*)alloc((size_t)n_nodes * F_HID * 4);
  float* h2    = (float*)alloc((size_t)n_nodes * F_HID * 4);

  const int TB = 256;
  auto layer = [&](const float* Xin, int K, const float* W, const float* q,
                   const float* k, const float* b, float* hout) {
    dim3 g(n_nodes / 16, R_REL);
    xform_wmma<<<g, 128, 0, stream>>>(Xin, W, xw, K);
    int totNR = n_nodes * R_REL;
    qk_kernel<<<(totNR + TB - 1) / TB, TB, 0, stream>>>(xw, q, k, qn, kn, totNR);
    init_amax<<<(n_nodes + TB - 1) / TB, TB, 0, stream>>>(amaxI, n_nodes);
    hipMemsetAsync(denom, 0, (size_t)n_nodes * 4, stream);
    hipMemsetAsync(hout, 0, (size_t)n_nodes * F_HID * 4, stream);
    edge_alpha<<<(n_edges + TB - 1) / TB, TB, 0, stream>>>(src, dst, etype, qn, kn,
                                                           exbuf, amaxI, n_edges);
    decode_amax<<<(n_nodes + TB - 1) / TB, TB, 0, stream>>>(amaxI, amaxF, n_nodes);
    edge_exp<<<(n_edges + TB - 1) / TB, TB, 0, stream>>>(dst, exbuf, amaxF, denom, n_edges);
    long long tot = (long long)n_edges * 32;
    edge_aggregate<<<(int)((tot + TB - 1) / TB), TB, 0, stream>>>(src, dst, etype, exbuf,
                                                                  denom, xw, hout, n_edges);
    int totH = n_nodes * F_HID;
    bias_relu<<<(totH + TB - 1) / TB, TB, 0, stream>>>(hout, b, totH);
  };

  layer(x,  F_IN,  w1, q1, k1, b1, h1);
  layer(h1, F_HID, w2, q2, k2, b2, h2);
  head_logsoftmax<<<(n_nodes + TB - 1) / TB, TB, 0, stream>>>(h2, lw, lb, out, n_nodes);
}